// GCNII_56642028700079
// MI455X (gfx1250) — compile-verified
//
#include <hip/hip_runtime.h>
#include <cmath>
#include <cstdint>

// ---------------------------------------------------------------------------
// GCNII on gfx1250: fp32 WMMA (V_WMMA_F32_16X16X4_F32) for all dense GEMMs,
// TDM (tensor_load_to_lds) for activation-tile staging, L2-resident fp32
// atomic SpMM for propagation.
// ---------------------------------------------------------------------------

typedef __attribute__((ext_vector_type(2))) float v2f;
typedef __attribute__((ext_vector_type(8))) float v8f;
typedef __attribute__((ext_vector_type(4))) unsigned int u32x4;
typedef __attribute__((ext_vector_type(4))) int i32x4;
typedef __attribute__((ext_vector_type(8))) int i32x8;

#define ALPHA_C 0.1f

__device__ __forceinline__ v8f wmma4(v2f a, v2f b, v8f c) {
  // D(16x16,f32) = A(16x4,f32) * B(4x16,f32) + C
  return __builtin_amdgcn_wmma_f32_16x16x4_f32(
      false, a, false, b, (short)0, c, false, false);
}

// TDM 2D tile load: copies tile_d0 x tile_d1 fp32 elements from a row-major
// tensor (row stride = stride0 elements) at gaddr into LDS at lds_byte_addr,
// inserting 2 pad DWORDs after every 64 DWORDs (matches [rows][66] layout).
// OOB rows (>= tensor_d1) read as zero.  Issue from ONE thread only.
__device__ __forceinline__ void tdm_load_tile_f32(
    unsigned int lds_byte_addr, const float* gaddr,
    unsigned int tensor_d0, unsigned int tensor_d1,
    unsigned int tile_d0, unsigned int tile_d1, unsigned int stride0)
{
  unsigned long long ga = (unsigned long long)(uintptr_t)gaddr;
  u32x4 g0;
  g0[0] = 1u;                                     // count=1, user descriptor
  g0[1] = lds_byte_addr;                          // LDS destination
  g0[2] = (unsigned int)ga;                       // global_addr[31:0]
  g0[3] = (unsigned int)((ga >> 32) & 0x1FFFFFFu) // global_addr[56:32]
        | (2u << 30);                             // type = 2 ("image")
  i32x8 g1;
  // data_size=4B(2), pad_enable, pad_interval=64 DW (code 5), pad_amount=2 DW (code 1)
  g1[0] = (int)((2u << 16) | (1u << 20) | (5u << 22) | (1u << 25));
  g1[1] = (int)((tensor_d0 & 0xFFFFu) << 16);                        // dim0[15:0]
  g1[2] = (int)(((tensor_d0 >> 16) & 0xFFFFu)                        // dim0[31:16]
              | ((tensor_d1 & 0xFFFFu) << 16));                      // dim1[15:0]
  g1[3] = (int)(((tensor_d1 >> 16) & 0xFFFFu)                        // dim1[31:16]
              | ((tile_d0 & 0xFFFFu) << 16));                        // tile_dim0
  g1[4] = (int)(tile_d1 & 0xFFFFu);                                  // tile_dim1 (0 => 1D)
  g1[5] = (int)stride0;                                              // dim0_stride[31:0]
  g1[6] = 0;
  g1[7] = 0;
  i32x4 z4 = {0, 0, 0, 0};               // groups 2/3: higher dims unused
  i32x8 z8 = {0, 0, 0, 0, 0, 0, 0, 0};   // extra descriptor words (unused)
  __builtin_amdgcn_tensor_load_to_lds(g0, g1, z4, z4, z8, 0);
}

// ----------------------------------------------------------------------------
// Kernel A: h0 = relu(x @ W0 + b0),  x:[N,256], W0:[256,64]
// 128 threads = 4 waves; block owns 64 rows; wave owns 16 rows x 64 cols.
// K processed in 64-wide chunks; x-tile staged by TDM, W-tile transposed by VALU.
// ----------------------------------------------------------------------------
__global__ void __launch_bounds__(128)
k_input_gemm(const float* __restrict__ x, const float* __restrict__ W0,
             const float* __restrict__ b0, float* __restrict__ out, int N)
{
  __shared__ float sX[64][66];   // 64 rows x 64-K chunk (stride 66, 8B aligned)
  __shared__ float sW[64][66];   // transposed weights: sW[n][k]

  const int tid  = threadIdx.x;
  const int wave = tid >> 5;
  const int lane = tid & 31;
  const int lo   = lane & 15;    // A: M index / B: N index
  const int hi   = lane >> 4;    // K-pair select (K = 2*hi, 2*hi+1)
  const int rowbase = blockIdx.x * 64;
  const unsigned int rows_left = (unsigned int)(N - rowbase);

  v8f acc[4] = {};

  for (int kc = 0; kc < 256; kc += 64) {
    __syncthreads();
    // async DMA: 64x64 fp32 sub-tile of x, padded to match sX[64][66]
    if (tid == 0) {
      tdm_load_tile_f32((unsigned int)(uintptr_t)&sX[0][0],
                        x + (size_t)rowbase * 256 + kc,
                        /*tensor_d0=*/64, /*tensor_d1=*/rows_left,
                        /*tile_d0=*/64, /*tile_d1=*/64, /*stride0=*/256);
      __builtin_amdgcn_s_wait_tensorcnt(0);
    }
    // stage W0 chunk transposed: sW[n][k] = W0[kc+k][n]
    for (int i = tid; i < 1024; i += 128) {
      int k = i >> 4, n4 = (i & 15) << 2;
      float4 v = *(const float4*)(W0 + (size_t)(kc + k) * 64 + n4);
      sW[n4 + 0][k] = v.x; sW[n4 + 1][k] = v.y;
      sW[n4 + 2][k] = v.z; sW[n4 + 3][k] = v.w;
    }
    __syncthreads();

    #pragma unroll
    for (int k0 = 0; k0 < 64; k0 += 4) {
      // A fragment: lane(lo,hi) holds A[lo][k0+2*hi], A[lo][k0+2*hi+1]
      v2f a = *(const v2f*)&sX[wave * 16 + lo][k0 + 2 * hi];
      #pragma unroll
      for (int n = 0; n < 4; ++n) {
        v2f b = *(const v2f*)&sW[n * 16 + lo][k0 + 2 * hi];
        acc[n] = wmma4(a, b, acc[n]);
      }
    }
  }

  // epilogue: bias + relu.  C/D layout: VGPR v -> row v (lanes 0-15) / v+8 (16-31)
  #pragma unroll
  for (int n = 0; n < 4; ++n) {
    int col = n * 16 + lo;
    float bias = b0[col];
    #pragma unroll
    for (int v = 0; v < 8; ++v) {
      int row = rowbase + wave * 16 + v + 8 * hi;
      if (row < N) {
        float val = acc[n][v] + bias;
        out[(size_t)row * 64 + col] = val > 0.f ? val : 0.f;
      }
    }
  }
}

// ----------------------------------------------------------------------------
// Kernel B: SpMM  agg[dst] += w * h[src]   (edge-parallel, fp32 L2 atomics)
// ----------------------------------------------------------------------------
__global__ void __launch_bounds__(256)
k_spmm(const float* __restrict__ h, const int* __restrict__ src,
       const int* __restrict__ dst, const float* __restrict__ w,
       float* __restrict__ agg, int E)
{
  int e = blockIdx.x * 256 + threadIdx.x;
  if (e >= E) return;
  int s = src[e], d = dst[e];
  float we = w[e];
  const float4* hs = (const float4*)(h + (size_t)s * 64);
  float* ad = agg + (size_t)d * 64;
  #pragma unroll
  for (int c = 0; c < 16; ++c) {
    float4 v = hs[c];
    __hip_atomic_fetch_add(ad + 4 * c + 0, v.x * we, __ATOMIC_RELAXED, __HIP_MEMORY_SCOPE_AGENT);
    __hip_atomic_fetch_add(ad + 4 * c + 1, v.y * we, __ATOMIC_RELAXED, __HIP_MEMORY_SCOPE_AGENT);
    __hip_atomic_fetch_add(ad + 4 * c + 2, v.z * we, __ATOMIC_RELAXED, __HIP_MEMORY_SCOPE_AGENT);
    __hip_atomic_fetch_add(ad + 4 * c + 3, v.w * we, __ATOMIC_RELAXED, __HIP_MEMORY_SCOPE_AGENT);
  }
}

// ----------------------------------------------------------------------------
// Kernel C: h = relu((1-beta)*S + beta*(S @ Wl)),  S = (1-alpha)*agg + alpha*x0
// (support tile needs arithmetic on two streams, so it is staged by VALU)
// ----------------------------------------------------------------------------
__global__ void __launch_bounds__(128)
k_layer(const float* __restrict__ agg, const float* __restrict__ x0,
        const float* __restrict__ Wl, float* __restrict__ out,
        float beta, int N)
{
  __shared__ float sS[64][66];   // support tile
  __shared__ float sW[64][66];   // transposed weights sW[n][k]

  const int tid  = threadIdx.x;
  const int wave = tid >> 5;
  const int lane = tid & 31;
  const int lo   = lane & 15;
  const int hi   = lane >> 4;
  const int rowbase = blockIdx.x * 64;

  for (int i = tid; i < 1024; i += 128) {
    int r = i >> 4, c4 = (i & 15) << 2;
    int gr = rowbase + r;
    float4 a = make_float4(0.f, 0.f, 0.f, 0.f);
    float4 xv = a;
    if (gr < N) {
      a  = *(const float4*)(agg + (size_t)gr * 64 + c4);
      xv = *(const float4*)(x0  + (size_t)gr * 64 + c4);
    }
    sS[r][c4 + 0] = (1.f - ALPHA_C) * a.x + ALPHA_C * xv.x;
    sS[r][c4 + 1] = (1.f - ALPHA_C) * a.y + ALPHA_C * xv.y;
    sS[r][c4 + 2] = (1.f - ALPHA_C) * a.z + ALPHA_C * xv.z;
    sS[r][c4 + 3] = (1.f - ALPHA_C) * a.w + ALPHA_C * xv.w;
  }
  for (int i = tid; i < 1024; i += 128) {
    int k = i >> 4, n4 = (i & 15) << 2;
    float4 v = *(const float4*)(Wl + (size_t)k * 64 + n4);
    sW[n4 + 0][k] = v.x; sW[n4 + 1][k] = v.y;
    sW[n4 + 2][k] = v.z; sW[n4 + 3][k] = v.w;
  }
  __syncthreads();

  v8f acc[4] = {};
  #pragma unroll
  for (int k0 = 0; k0 < 64; k0 += 4) {
    v2f a = *(const v2f*)&sS[wave * 16 + lo][k0 + 2 * hi];
    #pragma unroll
    for (int n = 0; n < 4; ++n) {
      v2f b = *(const v2f*)&sW[n * 16 + lo][k0 + 2 * hi];
      acc[n] = wmma4(a, b, acc[n]);
    }
  }

  #pragma unroll
  for (int n = 0; n < 4; ++n) {
    int col = n * 16 + lo;
    #pragma unroll
    for (int v = 0; v < 8; ++v) {
      int lr = wave * 16 + v + 8 * hi;
      int gr = rowbase + lr;
      if (gr < N) {
        float s = sS[lr][col];
        float val = (1.f - beta) * s + beta * acc[n][v];
        out[(size_t)gr * 64 + col] = val > 0.f ? val : 0.f;
      }
    }
  }
}

// ----------------------------------------------------------------------------
// Kernel D: out = log_softmax(h @ W1 + b1)  (64 classes, fused row-LSE)
// h tile is 64 contiguous rows x 64 cols = one linear TDM copy (with padding).
// ----------------------------------------------------------------------------
__global__ void __launch_bounds__(128)
k_output(const float* __restrict__ h, const float* __restrict__ W1,
         const float* __restrict__ b1, float* __restrict__ out, int N)
{
  __shared__ float sH[64][66];
  __shared__ float sW[64][66];
  __shared__ float sL[64][66];   // logits tile

  const int tid  = threadIdx.x;
  const int wave = tid >> 5;
  const int lane = tid & 31;
  const int lo   = lane & 15;
  const int hi   = lane >> 4;
  const int rowbase = blockIdx.x * 64;

  // async DMA: 64 rows of h (contiguous 4096 fp32) -> sH with row padding.
  // 1D descriptor; tensor_d0 = remaining elements so tail rows read as zero.
  if (tid == 0) {
    unsigned int rem = (unsigned int)((size_t)N * 64 - (size_t)rowbase * 64);
    tdm_load_tile_f32((unsigned int)(uintptr_t)&sH[0][0],
                      h + (size_t)rowbase * 64,
                      /*tensor_d0=*/rem, /*tensor_d1=*/1,
                      /*tile_d0=*/4096, /*tile_d1=*/0, /*stride0=*/4096);
    __builtin_amdgcn_s_wait_tensorcnt(0);
  }
  for (int i = tid; i < 1024; i += 128) {
    int k = i >> 4, n4 = (i & 15) << 2;
    float4 v = *(const float4*)(W1 + (size_t)k * 64 + n4);
    sW[n4 + 0][k] = v.x; sW[n4 + 1][k] = v.y;
    sW[n4 + 2][k] = v.z; sW[n4 + 3][k] = v.w;
  }
  __syncthreads();

  v8f acc[4] = {};
  #pragma unroll
  for (int k0 = 0; k0 < 64; k0 += 4) {
    v2f a = *(const v2f*)&sH[wave * 16 + lo][k0 + 2 * hi];
    #pragma unroll
    for (int n = 0; n < 4; ++n) {
      v2f b = *(const v2f*)&sW[n * 16 + lo][k0 + 2 * hi];
      acc[n] = wmma4(a, b, acc[n]);
    }
  }

  #pragma unroll
  for (int n = 0; n < 4; ++n) {
    int col = n * 16 + lo;
    float bias = b1[col];
    #pragma unroll
    for (int v = 0; v < 8; ++v) {
      sL[wave * 16 + v + 8 * hi][col] = acc[n][v] + bias;
    }
  }
  __syncthreads();

  if (tid < 64) {
    int gr = rowbase + tid;
    if (gr < N) {
      float m = sL[tid][0];
      #pragma unroll 8
      for (int c = 1; c < 64; ++c) m = fmaxf(m, sL[tid][c]);
      float sum = 0.f;
      #pragma unroll 8
      for (int c = 0; c < 64; ++c) sum += expf(sL[tid][c] - m);
      float lse = m + logf(sum);
      #pragma unroll 8
      for (int c = 0; c < 64; ++c)
        out[(size_t)gr * 64 + c] = sL[tid][c] - lse;
    }
  }
}

// ----------------------------------------------------------------------------
// Host launcher
// ----------------------------------------------------------------------------
extern "C" void kernel_launch(void* const* d_in, const int* in_sizes, int n_in,
                              void* d_out, int out_size, void* d_ws, size_t ws_size,
                              hipStream_t stream)
{
  (void)n_in; (void)out_size; (void)ws_size;
  const float* x  = (const float*)d_in[0];
  const int*   ei = (const int*)d_in[1];   // [2, E]: src row then dst row
  const float* ew = (const float*)d_in[2];
  const float* W0 = (const float*)d_in[3];
  const float* b0 = (const float*)d_in[4];
  const float* W1 = (const float*)d_in[5];
  const float* b1 = (const float*)d_in[6];
  const float* cW = (const float*)d_in[7]; // [8, 64, 64]

  const int N = in_sizes[0] / 256;
  const int E = in_sizes[2];

  float* x0  = (float*)d_ws;                 // [N,64] initial residual
  float* h   = x0 + (size_t)N * 64;          // [N,64] current features
  float* agg = h  + (size_t)N * 64;          // [N,64] SpMM accumulator

  const int nb = (N + 63) / 64;

  k_input_gemm<<<nb, 128, 0, stream>>>(x, W0, b0, x0, N);

  const float* hcur = x0;
  for (int l = 0; l < 8; ++l) {
    (void)hipMemsetAsync(agg, 0, (size_t)N * 64 * sizeof(float), stream);
    k_spmm<<<(E + 255) / 256, 256, 0, stream>>>(hcur, ei, ei + E, ew, agg, E);
    float beta = logf(0.5f / (float)(l + 1) + 1.0f);
    k_layer<<<nb, 128, 0, stream>>>(agg, x0, cW + (size_t)l * 64 * 64, h, beta, N);
    hcur = h;
  }

  k_output<<<nb, 128, 0, stream>>>(hcur, W1, b1, (float*)d_out, N);
}